// GraphConvLayer_74174085201985
// MI455X (gfx1250) — compile-verified
//
#include <hip/hip_runtime.h>
#include <math.h>

typedef __attribute__((ext_vector_type(2))) float v2f;
typedef __attribute__((ext_vector_type(8))) float v8f;

#define DIM 128
#define NT 8            // 8 column tiles of 16 -> 128 output cols
#define BN_EPS 1e-3f

// ---------------------------------------------------------------------------
// Prep: fold BatchNorm into weights.
//   s[k] = gamma[k]*rsqrt(var[k]+eps);  t[k] = beta[k] - mean[k]*s[k]
//   W'[k][n] = s[k]*W[k][n];            b'[n] = b[n] + sum_k t[k]*W[k][n]
// W' stored packed by K-pairs:  Wp[(k>>1)*256 + n*2 + (k&1)]
// so a WMMA B-fragment (K, K+1 for one column) is one contiguous b64 load.
// ---------------------------------------------------------------------------
__global__ __launch_bounds__(128) void prep_kernel(
    const float* __restrict__ g1, const float* __restrict__ be1,
    const float* __restrict__ m1, const float* __restrict__ v1,
    const float* __restrict__ W1, const float* __restrict__ b1,
    const float* __restrict__ g2, const float* __restrict__ be2,
    const float* __restrict__ m2, const float* __restrict__ v2,
    const float* __restrict__ W2, const float* __restrict__ b2,
    float* __restrict__ W1p, float* __restrict__ b1p,
    float* __restrict__ W2p, float* __restrict__ b2p)
{
    const int n = threadIdx.x;            // output column 0..127
    float acc = b1[n];
    for (int k = 0; k < DIM; ++k) {
        float s = g1[k] * rsqrtf(v1[k] + BN_EPS);
        float t = be1[k] - m1[k] * s;
        float w = W1[k * DIM + n];
        acc += t * w;
        W1p[(k >> 1) * 256 + n * 2 + (k & 1)] = s * w;
    }
    b1p[n] = acc;

    acc = b2[n];
    for (int k = 0; k < 2 * DIM; ++k) {
        float s = g2[k] * rsqrtf(v2[k] + BN_EPS);
        float t = be2[k] - m2[k] * s;
        float w = W2[k * DIM + n];
        acc += t * w;
        W2p[(k >> 1) * 256 + n * 2 + (k & 1)] = s * w;
    }
    b2p[n] = acc;
}

__global__ void zero_kernel(float* __restrict__ p, long n)
{
    long i = (long)blockIdx.x * blockDim.x + threadIdx.x;
    long stride = (long)gridDim.x * blockDim.x;
    for (; i < n; i += stride) p[i] = 0.0f;
}

__device__ __forceinline__ float gelu_exact(float x)
{
    return 0.5f * x * (1.0f + erff(x * 0.70710678118654752440f));
}

// ---------------------------------------------------------------------------
// Edge kernel: per wave, one tile of 16 edges.
//   msg[16x128] = gelu( x[nbr] @ W1' + b1' )   via v_wmma_f32_16x16x4_f32
//   atomicAdd into sums[node], atomicAdd 1 into cnt[node]
// ---------------------------------------------------------------------------
__global__ __launch_bounds__(128) void edge_kernel(
    const float* __restrict__ x,
    const int*   __restrict__ edges,   // [2,E]: [0..E)=dst node, [E..2E)=neighbour
    int E, int nTiles,
    const float* __restrict__ W1p, const float* __restrict__ b1p,
    float* __restrict__ sums, float* __restrict__ cnt)
{
    extern __shared__ float lW[];      // 64 KB packed W1'
    const int tid = threadIdx.x;
    {
        const float4* src = (const float4*)W1p;
        float4* dst = (float4*)lW;
        #pragma unroll 4
        for (int i = tid; i < DIM * DIM / 4; i += 128) dst[i] = src[i];
    }
    __syncthreads();

    const int wave = tid >> 5;
    const int lane = tid & 31;
    const int tile = blockIdx.x * 4 + wave;
    if (tile >= nTiles) return;

    const int ebase = tile * 16;
    const int r  = lane & 15;          // A row / B,C column within tile
    const int hi = lane >> 4;          // half-wave: K offset +2, M offset +8

    const int nbr = edges[E + ebase + r];
    const float* arow = x + (long)nbr * DIM + hi * 2;

    v8f acc[NT];
    #pragma unroll
    for (int t = 0; t < NT; ++t) acc[t] = (v8f)0.0f;

    #pragma unroll 2
    for (int kc = 0; kc < DIM / 4; ++kc) {
        const int k0 = kc * 4;
        v2f a = *(const v2f*)(arow + k0);               // K = k0+2*hi, +1
        const float* bb = lW + ((k0 >> 1) + hi) * 256 + r * 2;
        #pragma unroll
        for (int t = 0; t < NT; ++t) {
            v2f b = *(const v2f*)(bb + t * 32);
            acc[t] = __builtin_amdgcn_wmma_f32_16x16x4_f32(
                false, a, false, b, (short)0, acc[t], false, false);
        }
    }

    float bias[NT];
    #pragma unroll
    for (int t = 0; t < NT; ++t) bias[t] = b1p[t * 16 + r];

    #pragma unroll
    for (int j = 0; j < 8; ++j) {
        const int row = j + 8 * hi;                    // edge row in tile
        const int dst = edges[ebase + row];            // destination node
        float* drow = sums + (long)dst * DIM;
        #pragma unroll
        for (int t = 0; t < NT; ++t) {
            float v = gelu_exact(acc[t][j] + bias[t]);
            atomicAdd(drow + t * 16 + r, v);
        }
    }
    if (lane < 16) atomicAdd(cnt + edges[ebase + lane], 1.0f);
}

// ---------------------------------------------------------------------------
// Node kernel: per wave, one tile of 16 nodes.
//   h = [x | sums*inv_cnt]  (K=256),  out = gelu( h @ W2' + b2' )
// ---------------------------------------------------------------------------
__global__ __launch_bounds__(128) void node_kernel(
    const float* __restrict__ x,
    const float* __restrict__ W2p, const float* __restrict__ b2p,
    const float* __restrict__ sums, const float* __restrict__ cnt,
    float* __restrict__ out, int nTiles)
{
    extern __shared__ float lW[];      // 128 KB packed W2'
    const int tid = threadIdx.x;
    {
        const float4* src = (const float4*)W2p;
        float4* dst = (float4*)lW;
        #pragma unroll 4
        for (int i = tid; i < 2 * DIM * DIM / 4; i += 128) dst[i] = src[i];
    }
    __syncthreads();

    const int wave = tid >> 5;
    const int lane = tid & 31;
    const int tile = blockIdx.x * 4 + wave;
    if (tile >= nTiles) return;

    const int r  = lane & 15;
    const int hi = lane >> 4;
    const int node = tile * 16 + r;

    const float invc = 1.0f / fmaxf(cnt[node], 1.0f);
    const float* xrow = x    + (long)node * DIM + hi * 2;
    const float* srow = sums + (long)node * DIM + hi * 2;

    v8f acc[NT];
    #pragma unroll
    for (int t = 0; t < NT; ++t) acc[t] = (v8f)0.0f;

    // K in [0,128): raw node features
    #pragma unroll 2
    for (int kc = 0; kc < DIM / 4; ++kc) {
        const int k0 = kc * 4;
        v2f a = *(const v2f*)(xrow + k0);
        const float* bb = lW + ((k0 >> 1) + hi) * 256 + r * 2;
        #pragma unroll
        for (int t = 0; t < NT; ++t) {
            v2f b = *(const v2f*)(bb + t * 32);
            acc[t] = __builtin_amdgcn_wmma_f32_16x16x4_f32(
                false, a, false, b, (short)0, acc[t], false, false);
        }
    }
    // K in [128,256): aggregated means (sums * inv_cnt)
    #pragma unroll 2
    for (int kc = 0; kc < DIM / 4; ++kc) {
        const int k0 = kc * 4;
        v2f s_ = *(const v2f*)(srow + k0);
        v2f a; a.x = s_.x * invc; a.y = s_.y * invc;
        const float* bb = lW + (((k0 + DIM) >> 1) + hi) * 256 + r * 2;
        #pragma unroll
        for (int t = 0; t < NT; ++t) {
            v2f b = *(const v2f*)(bb + t * 32);
            acc[t] = __builtin_amdgcn_wmma_f32_16x16x4_f32(
                false, a, false, b, (short)0, acc[t], false, false);
        }
    }

    float bias[NT];
    #pragma unroll
    for (int t = 0; t < NT; ++t) bias[t] = b2p[t * 16 + r];

    #pragma unroll
    for (int j = 0; j < 8; ++j) {
        const int row = j + 8 * hi;
        float* orow = out + (long)(tile * 16 + row) * DIM;
        #pragma unroll
        for (int t = 0; t < NT; ++t) {
            orow[t * 16 + r] = gelu_exact(acc[t][j] + bias[t]);
        }
    }
}

// ---------------------------------------------------------------------------
extern "C" void kernel_launch(void* const* d_in, const int* in_sizes, int n_in,
                              void* d_out, int out_size, void* d_ws, size_t ws_size,
                              hipStream_t stream)
{
    const float* x     = (const float*)d_in[0];   // node_repr [N,128]
    const int*   edges = (const int*)  d_in[1];   // [2,E]
    const float* g1 = (const float*)d_in[2];
    const float* be1= (const float*)d_in[3];
    const float* m1 = (const float*)d_in[4];
    const float* v1 = (const float*)d_in[5];
    const float* W1 = (const float*)d_in[6];
    const float* b1 = (const float*)d_in[7];
    const float* g2 = (const float*)d_in[8];
    const float* be2= (const float*)d_in[9];
    const float* m2 = (const float*)d_in[10];
    const float* v2 = (const float*)d_in[11];
    const float* W2 = (const float*)d_in[12];
    const float* b2 = (const float*)d_in[13];

    const int N = in_sizes[0] / DIM;
    const int E = in_sizes[1] / 2;

    char* ws = (char*)d_ws;
    float* W1p  = (float*)(ws);                       //  64 KB
    float* b1p  = (float*)(ws + 65536);               //  512 B
    float* W2p  = (float*)(ws + 66048);               // 128 KB
    float* b2p  = (float*)(ws + 197120);              //  512 B
    float* sums = (float*)(ws + 197632);              // N*128*4 B
    float* cntp = sums + (long)N * DIM;               // N*4 B

    prep_kernel<<<1, 128, 0, stream>>>(g1, be1, m1, v1, W1, b1,
                                       g2, be2, m2, v2, W2, b2,
                                       W1p, b1p, W2p, b2p);

    zero_kernel<<<2048, 256, 0, stream>>>(sums, (long)N * DIM + N);

    const int eTiles = (E + 15) / 16;
    edge_kernel<<<(eTiles + 3) / 4, 128, DIM * DIM * 4, stream>>>(
        x, edges, E, eTiles, W1p, b1p, sums, cntp);

    const int nTiles = (N + 15) / 16;
    node_kernel<<<(nTiles + 3) / 4, 128, 2 * DIM * DIM * 4, stream>>>(
        x, W2p, b2p, sums, cntp, (float*)d_out, nTiles);
}